// ISSMDecoderLayer_67044439491007
// MI455X (gfx1250) — compile-verified
//
#include <hip/hip_runtime.h>
#include <math.h>
#include <stdint.h>

// ---- problem constants (from reference) ----
#define B_    2
#define L_    1024
#define K_    128
#define D_    256
#define DI_   512
#define H_    8
#define HD_   64
#define CC_   514      // DI + 2*NGROUPS
#define ZDIM  1034     // 2*DI + 2*G + H
#define DTOFF 1026     // 2*DI + 2*G
#define EPS_  1e-5f

typedef __attribute__((ext_vector_type(2))) float v2f;
typedef __attribute__((ext_vector_type(8))) float v8f;
typedef __attribute__((ext_vector_type(4))) unsigned int u32x4;
typedef __attribute__((ext_vector_type(8))) unsigned int u32x8;

__device__ __forceinline__ float siluf(float v) { return v / (1.f + __expf(-v)); }
__device__ __forceinline__ float softplusf(float v) { return v > 20.f ? v : log1pf(__expf(v)); }
__device__ __forceinline__ int clampi(int v, int lo, int hi) { return v < lo ? lo : (v > hi ? hi : v); }

// LDS byte offset of a __shared__ object (generic -> AS3 addrspacecast).
__device__ __forceinline__ uint32_t lds_off(const void* p)
{
    return (uint32_t)(uintptr_t)(__attribute__((address_space(3))) const char*)p;
}

// ============================================================
// WMMA fp32 GEMM:  Out(M,N) = A(M,K) * W(N,K)^T
// One wave per 16x16 output tile; V_WMMA_F32_16X16X4_F32.
// GUARD=true handles N not a multiple of 16 (zero-fill + store guard);
// GUARD=false is the branch-free exact-N path (no cndmask in hot loop).
// A (16x4/step): lane l -> M=l%16, K = 2*(l/16)+{0,1}
// B (4x16/step): lane l -> N=l%16, K = 2*(l/16)+{0,1}
// C/D:           lane l, vgpr v -> N=l%16, M = v + 8*(l/16)
// ============================================================
template <bool GUARD>
__global__ void k_wmma_gemm(const float* __restrict__ A, const float* __restrict__ W,
                            float* __restrict__ Out, int M, int N, int Kt,
                            int ldA, int ldW, int ldOut)
{
    int lane = threadIdx.x;                 // 0..31 (wave32)
    int n0 = blockIdx.x * 16;
    int m0 = blockIdx.y * 16;
    int half = lane >> 4;                   // 0/1
    int lm   = lane & 15;

    int arow = m0 + lm;                     // always < M
    int wrow = n0 + lm;
    bool wvalid = GUARD ? (wrow < N) : true;
    int wr = (GUARD && !wvalid) ? (N - 1) : wrow;

    const float* Ap = A + (size_t)arow * ldA + 2 * half;
    const float* Wp = W + (size_t)wr   * ldW + 2 * half;

    v8f acc = {};
    for (int k0 = 0; k0 < Kt; k0 += 4) {
        v2f a, b;
        a.x = Ap[k0];
        a.y = Ap[k0 + 1];
        float bx = Wp[k0];
        float by = Wp[k0 + 1];
        if (GUARD) {
            b.x = wvalid ? bx : 0.f;
            b.y = wvalid ? by : 0.f;
        } else {
            b.x = bx;
            b.y = by;
        }
        acc = __builtin_amdgcn_wmma_f32_16x16x4_f32(false, a, false, b,
                                                    (short)0, acc, false, false);
    }
    if (!GUARD || wvalid) {
        int col = n0 + lm;
        #pragma unroll
        for (int v = 0; v < 8; ++v) {
            int row = m0 + v + 8 * half;
            Out[(size_t)row * ldOut + col] = acc[v];
        }
    }
}

// WMMA GEMM for h0 = (in_query @ Wq^T), scatter-stored transposed into
// h0t[b][h][d][k] so the scan reads contiguously. M=256 (b*K+k), N=512 (h*64+d), K=256.
__global__ void k_wmma_gemm_h0(const float* __restrict__ A, const float* __restrict__ W,
                               float* __restrict__ h0t)
{
    int lane = threadIdx.x;
    int n0 = blockIdx.x * 16;
    int m0 = blockIdx.y * 16;
    int half = lane >> 4;
    int lm   = lane & 15;

    const float* Ap = A + (size_t)(m0 + lm) * D_ + 2 * half;
    const float* Wp = W + (size_t)(n0 + lm) * D_ + 2 * half;

    v8f acc = {};
    for (int k0 = 0; k0 < D_; k0 += 4) {
        v2f a, b;
        a.x = Ap[k0];  a.y = Ap[k0 + 1];
        b.x = Wp[k0];  b.y = Wp[k0 + 1];
        acc = __builtin_amdgcn_wmma_f32_16x16x4_f32(false, a, false, b,
                                                    (short)0, acc, false, false);
    }
    int col = n0 + lm;          // h*64 + d
    int h = col >> 6, dd = col & 63;
    #pragma unroll
    for (int v = 0; v < 8; ++v) {
        int row = m0 + v + 8 * half;   // b*128 + k
        int b = row >> 7, k = row & 127;
        h0t[(((size_t)(b * H_ + h)) * HD_ + dd) * K_ + k] = acc[v];
    }
}

// ============================================================
// Depthwise conv (kernel=7, pad=3) along L + SiLU.
// ============================================================
__global__ void k_dwconv_silu(const float* __restrict__ in, int inStride,
                              const float* __restrict__ w, const float* __restrict__ bias,
                              float* __restrict__ out)
{
    int idx = blockIdx.x * blockDim.x + threadIdx.x;       // B*L*CC
    if (idx >= B_ * L_ * CC_) return;
    int c  = idx % CC_;
    int bl = idx / CC_;
    int l  = bl % L_;
    int b  = bl / L_;
    float acc = bias[c];
    #pragma unroll
    for (int t = 0; t < 7; ++t) {
        int ll = l - 3 + t;
        if (ll >= 0 && ll < L_)
            acc = fmaf(in[(size_t)(b * L_ + ll) * inStride + c], w[c * 7 + t], acc);
    }
    out[(size_t)bl * CC_ + c] = siluf(acc);
}

// ============================================================
// table[10][10][10][4] = relu(coords @ w1^T + b1) @ w2^T
// ============================================================
__global__ void k_table(const float* __restrict__ w1, const float* __restrict__ b1,
                        const float* __restrict__ w2, float* __restrict__ table)
{
    int idx = blockIdx.x * blockDim.x + threadIdx.x;
    if (idx >= 1000) return;
    int a = idx / 100, b = (idx / 10) % 10, c = idx % 10;
    float px = -4.f + 8.f * a / 9.f;
    float py = -4.f + 8.f * b / 9.f;
    float pz = -4.f + 8.f * c / 9.f;
    float o0 = 0.f, o1 = 0.f, o2 = 0.f, o3 = 0.f;
    for (int j = 0; j < 128; ++j) {
        float hd = fmaf(w1[j * 3], px, fmaf(w1[j * 3 + 1], py, fmaf(w1[j * 3 + 2], pz, b1[j])));
        hd = fmaxf(hd, 0.f);
        o0 = fmaf(w2[j],       hd, o0);
        o1 = fmaf(w2[128 + j], hd, o1);
        o2 = fmaf(w2[256 + j], hd, o2);
        o3 = fmaf(w2[384 + j], hd, o3);
    }
    table[idx * 4 + 0] = o0; table[idx * 4 + 1] = o1;
    table[idx * 4 + 2] = o2; table[idx * 4 + 3] = o3;
}

// ============================================================
// Box corners + rotation matrices per (b,k).
// ============================================================
__global__ void k_corners(const float* __restrict__ qxyz, const float* __restrict__ qsize,
                          const float* __restrict__ qang,
                          float* __restrict__ corners, float* __restrict__ Rm)
{
    int idx = blockIdx.x * blockDim.x + threadIdx.x;   // B*K = 256
    if (idx >= B_ * K_) return;
    float cx = qxyz[idx * 3], cy = qxyz[idx * 3 + 1], cz = qxyz[idx * 3 + 2];
    float camc0 = cx, camc1 = -cz, camc2 = cy;
    float ang = qang[idx];
    float cs = cosf(ang), sn = sinf(ang);
    float R[9] = { cs, 0.f, sn,  0.f, 1.f, 0.f,  -sn, 0.f, cs };
    #pragma unroll
    for (int j = 0; j < 9; ++j) Rm[(size_t)idx * 9 + j] = R[j];

    const float sxA[8] = {1,1,-1,-1,1,1,-1,-1};
    const float syA[8] = {1,1,1,1,-1,-1,-1,-1};
    const float szA[8] = {1,-1,-1,1,1,-1,-1,1};
    float lsz = qsize[idx * 3], wsz = qsize[idx * 3 + 1], hsz = qsize[idx * 3 + 2];
    #pragma unroll
    for (int c = 0; c < 8; ++c) {
        float l0 = lsz * sxA[c] * 0.5f;
        float l1 = hsz * syA[c] * 0.5f;
        float l2 = wsz * szA[c] * 0.5f;
        float cam0 = R[0] * l0 + R[1] * l1 + R[2] * l2 + camc0;
        float cam1 = R[3] * l0 + R[4] * l1 + R[5] * l2 + camc1;
        float cam2 = R[6] * l0 + R[7] * l1 + R[8] * l2 + camc2;
        float* cp = corners + ((size_t)idx * 8 + c) * 3;
        cp[0] = cam0; cp[1] = cam2; cp[2] = -cam1;
    }
}

// ============================================================
// dist[b][l][k][4]: sum over 8 corners of trilinear(table, g(corner, key)).
// ============================================================
__device__ __forceinline__ float logcomp(float v)
{
    float s = (v > 0.f) ? 1.f : (v < 0.f ? -1.f : 0.f);
    return s * log2f(fabsf(v) * 512.f + 1.f) * (1.f / 3.f);
}

__global__ void k_dist(const float* __restrict__ key_xyz, const float* __restrict__ corners,
                       const float* __restrict__ Rm, const float* __restrict__ table,
                       float* __restrict__ dist)
{
    int idx = blockIdx.x * blockDim.x + threadIdx.x;  // B*L*K
    if (idx >= B_ * L_ * K_) return;
    int k = idx & 127;
    int l = (idx >> 7) & 1023;
    int b = idx >> 17;
    const float* kp = key_xyz + (size_t)(b * L_ + l) * 3;
    float kx = kp[0], ky = kp[1], kz = kp[2];

    float o0 = 0.f, o1 = 0.f, o2 = 0.f, o3 = 0.f;
    for (int c = 0; c < 8; ++c) {
        const float* cp = corners + ((size_t)(b * K_ + k) * 8 + c) * 3;
        float dx = cp[0] - kx, dy = cp[1] - ky, dz = cp[2] - kz;
        float e0 = dx, e1 = -dz, e2 = dy;
        // reference tiles angle as (1,8): flattened q = k*8+c uses angle[q % K]
        int q = (k * 8 + c) & (K_ - 1);
        const float* R = Rm + (size_t)(b * K_ + q) * 9;
        float f0 = e0 * R[0] + e1 * R[3] + e2 * R[6];
        float f1 = e0 * R[1] + e1 * R[4] + e2 * R[7];
        float f2 = e0 * R[2] + e1 * R[5] + e2 * R[8];
        float g0 = logcomp(f0)  * 0.25f;   //  d3[0]
        float g1 = logcomp(f2)  * 0.25f;   //  d3[2]
        float g2 = logcomp(-f1) * 0.25f;   // -d3[1]
        // trilinear: k<-g0, j<-g1, i<-g2
        float pk = ((g0 + 1.f) * 10.f - 1.f) * 0.5f;
        float pj = ((g1 + 1.f) * 10.f - 1.f) * 0.5f;
        float pi = ((g2 + 1.f) * 10.f - 1.f) * 0.5f;
        float i0f = floorf(pi), j0f = floorf(pj), k0f = floorf(pk);
        float wi = pi - i0f, wj = pj - j0f, wk = pk - k0f;
        int i0 = (int)i0f, j0 = (int)j0f, k0 = (int)k0f;
        #pragma unroll
        for (int a = 0; a < 2; ++a)
        #pragma unroll
        for (int bb = 0; bb < 2; ++bb)
        #pragma unroll
        for (int cc = 0; cc < 2; ++cc) {
            int ii = i0 + a, jj = j0 + bb, kk = k0 + cc;
            bool valid = (ii >= 0) & (ii < 10) & (jj >= 0) & (jj < 10) & (kk >= 0) & (kk < 10);
            const float* tv = table + (size_t)(clampi(ii,0,9) * 100 + clampi(jj,0,9) * 10 + clampi(kk,0,9)) * 4;
            float w = (a ? wi : 1.f - wi) * (bb ? wj : 1.f - wj) * (cc ? wk : 1.f - wk);
            w = valid ? w : 0.f;
            o0 = fmaf(w, tv[0], o0); o1 = fmaf(w, tv[1], o1);
            o2 = fmaf(w, tv[2], o2); o3 = fmaf(w, tv[3], o3);
        }
    }
    dist[(size_t)idx * 4 + 0] = o0; dist[(size_t)idx * 4 + 1] = o1;
    dist[(size_t)idx * 4 + 2] = o2; dist[(size_t)idx * 4 + 3] = o3;
}

// ============================================================
// Per (b,l,k): dA[b][l][h][k], dB fwd/bwd, C fwd/bwd.
// ============================================================
__global__ void k_bcdt(const float* __restrict__ dist, const float* __restrict__ zx,
                       const float* __restrict__ xf, const float* __restrict__ xb,
                       const float* __restrict__ Wbc, const float* __restrict__ Wdt,
                       const float* __restrict__ dt_bias, const float* __restrict__ A_log,
                       float* __restrict__ dA, float* __restrict__ dBf, float* __restrict__ dBb,
                       float* __restrict__ CmF, float* __restrict__ CmB)
{
    int idx = blockIdx.x * blockDim.x + threadIdx.x;  // B*L*K
    if (idx >= B_ * L_ * K_) return;
    int k  = idx & 127;
    int bl = idx >> 7;
    float d0 = dist[(size_t)idx * 4 + 0], d1 = dist[(size_t)idx * 4 + 1];
    float d2 = dist[(size_t)idx * 4 + 2], d3 = dist[(size_t)idx * 4 + 3];
    float Bm0 = d0 * Wbc[0] + d1 * Wbc[1] + d2 * Wbc[2] + d3 * Wbc[3];
    float Cm0 = d0 * Wbc[4] + d1 * Wbc[5] + d2 * Wbc[6] + d3 * Wbc[7];
    float bbF = xf[(size_t)bl * CC_ + DI_];
    float cbF = xf[(size_t)bl * CC_ + DI_ + 1];
    float bbB = xb[(size_t)bl * CC_ + DI_];
    float cbB = xb[(size_t)bl * CC_ + DI_ + 1];
    CmF[idx] = Cm0 + cbF;
    CmB[idx] = Cm0 + cbB;
    float BmF = Bm0 + bbF, BmB = Bm0 + bbB;
    #pragma unroll
    for (int h = 0; h < H_; ++h) {
        float t = d0 * Wdt[h * 4] + d1 * Wdt[h * 4 + 1] + d2 * Wdt[h * 4 + 2] + d3 * Wdt[h * 4 + 3]
                + zx[(size_t)bl * ZDIM + DTOFF + h] + dt_bias[h];
        float dt = softplusf(t);
        float a  = -__expf(A_log[h]);
        size_t o = ((size_t)bl * H_ + h) * K_ + k;
        dA[o]  = __expf(dt * a);
        dBf[o] = dt * BmF;
        dBb[o] = dt * BmB;
    }
}

// ============================================================
// Selective scan. 32 blocks = (dir, b, h). 256 threads = 8 waves.
// Thread: d = tid/4 (fixed), k in [(tid%4)*32, +32) -> 32 state f32 in VGPRs.
// Per-step staging: waves 0..2 issue GLOBAL_LOAD_ASYNC_TO_LDS_B128 (ASYNCcnt),
// wave 3 issues a TDM TENSOR_LOAD_TO_LDS with a 2-group D# (TENSORcnt).
// y reduced with 2 wave32 shuffles; next step prefetched (global_prefetch_b8).
// ============================================================
__global__ void k_scan(const float* __restrict__ dA,
                       const float* __restrict__ dBf, const float* __restrict__ dBb,
                       const float* __restrict__ CmF, const float* __restrict__ CmB,
                       const float* __restrict__ xf,  const float* __restrict__ xb,
                       const float* __restrict__ h0t,
                       float* __restrict__ yF, float* __restrict__ yB,
                       float* __restrict__ hF, float* __restrict__ hB)
{
    int blk = blockIdx.x;
    int dir = blk >> 4;
    int b   = (blk >> 3) & 1;
    int h   = blk & 7;
    int tid = threadIdx.x;
    int d   = tid >> 2;
    int kb  = (tid & 3) << 5;
    int wv  = tid >> 5;        // wave id 0..7
    int ln  = tid & 31;        // lane id

    const float* dB = dir ? dBb : dBf;
    const float* Cm = dir ? CmB : CmF;
    const float* xs = dir ? xb  : xf;
    float* y  = dir ? yB : yF;
    float* hO = dir ? hB : hF;

    __shared__ float s_dA[K_], s_dB[K_], s_C[K_], s_x[HD_];

    float hst[32];
    {
        const float* p = h0t + (((size_t)(b * H_ + h)) * HD_ + d) * K_ + kb;
        #pragma unroll
        for (int i = 0; i < 32; ++i) hst[i] = p[i];
    }

    for (int s = 0; s < L_; ++s) {
        int t = dir ? (L_ - 1 - s) : s;
        size_t bl   = (size_t)b * L_ + t;
        size_t base = (bl * H_ + h) * K_;
        if (s + 1 < L_) {
            int tn = dir ? (L_ - 2 - s) : (s + 1);
            size_t bn = (((size_t)b * L_ + tn) * H_ + h) * K_;
            __builtin_prefetch(dA + bn, 0, 1);
            __builtin_prefetch(dB + bn, 0, 1);
        }
        // ---- CDNA5 async staging into LDS ----
        if (wv == 0) {
            uint64_t g = (uint64_t)(uintptr_t)(dA + base + ln * 4);
            uint32_t l = lds_off(&s_dA[ln * 4]);
            asm volatile("global_load_async_to_lds_b128 %0, %1, off" :: "v"(l), "v"(g) : "memory");
        } else if (wv == 1) {
            uint64_t g = (uint64_t)(uintptr_t)(dB + base + ln * 4);
            uint32_t l = lds_off(&s_dB[ln * 4]);
            asm volatile("global_load_async_to_lds_b128 %0, %1, off" :: "v"(l), "v"(g) : "memory");
        } else if (wv == 2) {
            uint64_t g = (uint64_t)(uintptr_t)(Cm + bl * K_ + ln * 4);
            uint32_t l = lds_off(&s_C[ln * 4]);
            asm volatile("global_load_async_to_lds_b128 %0, %1, off" :: "v"(l), "v"(g) : "memory");
        } else if (wv == 3) {
            // TDM: 64x1 tile of 4-byte elements, global -> LDS (s_x)
            uint64_t ga = (uint64_t)(uintptr_t)(xs + bl * CC_ + h * HD_);
            u32x4 g0;
            g0[0] = 1u;                                             // count=1, user mode
            g0[1] = lds_off(&s_x[0]);                               // lds_addr (bytes)
            g0[2] = (uint32_t)ga;                                   // global_addr[31:0]
            g0[3] = (uint32_t)((ga >> 32) & 0x1FFFFFFu) | (2u << 30); // addr[56:32], type=2
            u32x8 g1;
            g1[0] = 2u << 16;      // workgroup_mask=0, data_size=2 (4B)
            g1[1] = 64u << 16;     // tensor_dim0 = 64 (bits 63:48)
            g1[2] = 1u << 16;      // tensor_dim0 hi = 0, tensor_dim1 = 1 (bits 95:80)
            g1[3] = 64u << 16;     // tensor_dim1 hi = 0, tile_dim0 = 64 (bits 127:112)
            g1[4] = 1u;            // tile_dim1 = 1, tile_dim2 = 0
            g1[5] = 64u;           // tensor_dim0_stride = 64 (lo)
            g1[6] = 64u << 16;     // dim0_stride hi = 0, tensor_dim1_stride lo = 64
            g1[7] = 0u;            // tensor_dim1_stride hi
            asm volatile("tensor_load_to_lds %0, %1" :: "s"(g0), "s"(g1) : "memory");
        }
        asm volatile("s_wait_asynccnt 0" ::: "memory");
        __builtin_amdgcn_s_wait_tensorcnt(0);
        __syncthreads();

        float xv  = s_x[d];
        float acc = 0.f;
        #pragma unroll
        for (int i = 0; i < 32; ++i) {
            int k = kb + i;
            float hv = fmaf(hst[i], s_dA[k], xv * s_dB[k]);
            hst[i] = hv;
            acc = fmaf(hv, s_C[k], acc);
        }
        acc += __shfl_xor(acc, 1);
        acc += __shfl_xor(acc, 2);
        if ((tid & 3) == 0) y[bl * (size_t)DI_ + h * HD_ + d] = acc;
        __syncthreads();
    }
    float* p = hO + (((size_t)(b * H_ + h)) * HD_ + d) * K_ + kb;
    #pragma unroll
    for (int i = 0; i < 32; ++i) p[i] = hst[i];
}

// ============================================================
// y = yF + yB + Dp*x; gated = y * silu(z); RMS-norm over 512 * key_norm_w
// ============================================================
__global__ void k_gate_norm(const float* __restrict__ yF, const float* __restrict__ yB,
                            const float* __restrict__ xf, const float* __restrict__ zx,
                            const float* __restrict__ Dp, const float* __restrict__ knw,
                            float* __restrict__ out)
{
    int bl  = blockIdx.x;       // B*L
    int tid = threadIdx.x;      // 256
    __shared__ float red[256];
    float g[2];
    float ssum = 0.f;
    #pragma unroll
    for (int j = 0; j < 2; ++j) {
        int c = tid * 2 + j;
        int h = c >> 6;
        float xv = xf[(size_t)bl * CC_ + c];
        float yv = yF[(size_t)bl * DI_ + c] + yB[(size_t)bl * DI_ + c] + Dp[h] * xv;
        float z  = zx[(size_t)bl * ZDIM + c];
        g[j] = yv * siluf(z);
        ssum = fmaf(g[j], g[j], ssum);
    }
    red[tid] = ssum;
    __syncthreads();
    for (int off = 128; off > 0; off >>= 1) {
        if (tid < off) red[tid] += red[tid + off];
        __syncthreads();
    }
    float r = rsqrtf(red[0] * (1.f / DI_) + EPS_);
    #pragma unroll
    for (int j = 0; j < 2; ++j) {
        int c = tid * 2 + j;
        out[(size_t)bl * DI_ + c] = g[j] * r * knw[c];
    }
}

// ============================================================
// state = 0.5*(hF+hB) gathered to (b,k,512); LayerNorm * ln_w + ln_b
// ============================================================
__global__ void k_state_norm(const float* __restrict__ hF, const float* __restrict__ hB,
                             const float* __restrict__ lnw, const float* __restrict__ lnb,
                             float* __restrict__ out)
{
    int bk  = blockIdx.x;     // B*K
    int b   = bk >> 7, k = bk & 127;
    int tid = threadIdx.x;    // 256
    __shared__ float rs[256], rq[256];
    float v[2];
    float s = 0.f, q = 0.f;
    #pragma unroll
    for (int j = 0; j < 2; ++j) {
        int c = tid * 2 + j;
        int h = c >> 6, d = c & 63;
        size_t idx = (((size_t)(b * H_ + h)) * HD_ + d) * K_ + k;
        v[j] = 0.5f * (hF[idx] + hB[idx]);
        s += v[j];
        q = fmaf(v[j], v[j], q);
    }
    rs[tid] = s; rq[tid] = q;
    __syncthreads();
    for (int off = 128; off > 0; off >>= 1) {
        if (tid < off) { rs[tid] += rs[tid + off]; rq[tid] += rq[tid + off]; }
        __syncthreads();
    }
    float mu  = rs[0] * (1.f / DI_);
    float var = rq[0] * (1.f / DI_) - mu * mu;
    float inv = rsqrtf(var + EPS_);
    #pragma unroll
    for (int j = 0; j < 2; ++j) {
        int c = tid * 2 + j;
        out[(size_t)bk * DI_ + c] = (v[j] - mu) * inv * lnw[c] + lnb[c];
    }
}

// ============================================================
// Host launcher
// ============================================================
extern "C" void kernel_launch(void* const* d_in, const int* in_sizes, int n_in,
                              void* d_out, int out_size, void* d_ws, size_t ws_size,
                              hipStream_t stream)
{
    (void)in_sizes; (void)n_in; (void)out_size; (void)ws_size;
    const float* in_key      = (const float*)d_in[0];
    const float* in_query    = (const float*)d_in[1];
    const float* key_xyz     = (const float*)d_in[2];
    const float* query_xyz   = (const float*)d_in[3];
    const float* query_size  = (const float*)d_in[4];
    const float* query_angle = (const float*)d_in[5];
    const float* Wkey        = (const float*)d_in[6];
    const float* conv_w      = (const float*)d_in[7];
    const float* conv_b      = (const float*)d_in[8];
    const float* convb_w     = (const float*)d_in[9];
    const float* convb_b     = (const float*)d_in[10];
    const float* Wq          = (const float*)d_in[11];
    const float* Wbc         = (const float*)d_in[12];
    const float* Wdt         = (const float*)d_in[13];
    const float* dt_bias     = (const float*)d_in[14];
    const float* A_log       = (const float*)d_in[15];
    const float* Dp          = (const float*)d_in[16];
    const float* Wok         = (const float*)d_in[17];
    const float* Woq         = (const float*)d_in[18];
    const float* key_norm_w  = (const float*)d_in[19];
    const float* ln_w        = (const float*)d_in[20];
    const float* ln_b        = (const float*)d_in[21];
    const float* cpb_w1      = (const float*)d_in[22];
    const float* cpb_b1      = (const float*)d_in[23];
    const float* cpb_w2      = (const float*)d_in[24];
    float* out = (float*)d_out;

    float* ws = (float*)d_ws;
    size_t off = 0;
    auto alloc = [&](size_t n) { float* p = ws + off; off += n; return p; };
    float* zx      = alloc((size_t)B_ * L_ * ZDIM);      // zxbcdt
    float* xf      = alloc((size_t)B_ * L_ * CC_);       // xbc_f
    float* xb      = alloc((size_t)B_ * L_ * CC_);       // xbc_b
    float* table   = alloc(1000 * 4);
    float* corners = alloc((size_t)B_ * K_ * 8 * 3);
    float* Rm      = alloc((size_t)B_ * K_ * 9);
    float* dist    = alloc((size_t)B_ * L_ * K_ * 4);
    float* dA      = alloc((size_t)B_ * L_ * H_ * K_);
    float* dBf     = alloc((size_t)B_ * L_ * H_ * K_);
    float* dBb     = alloc((size_t)B_ * L_ * H_ * K_);
    float* CmF     = alloc((size_t)B_ * L_ * K_);
    float* CmB     = alloc((size_t)B_ * L_ * K_);
    float* h0t     = alloc((size_t)B_ * H_ * HD_ * K_);
    float* hF      = alloc((size_t)B_ * H_ * HD_ * K_);
    float* hB      = alloc((size_t)B_ * H_ * HD_ * K_);
    float* yF      = alloc((size_t)B_ * L_ * DI_);
    float* yB      = alloc((size_t)B_ * L_ * DI_);
    float* gatedn  = alloc((size_t)B_ * L_ * DI_);
    float* stn     = alloc((size_t)B_ * K_ * DI_);

    // 1) zxbcdt = in_key @ Wkey^T   (2048 x 1034 x 256) -- N tail needs guard
    k_wmma_gemm<true><<<dim3((ZDIM + 15) / 16, (B_ * L_) / 16), 32, 0, stream>>>(
        in_key, Wkey, zx, B_ * L_, ZDIM, D_, D_, D_, ZDIM);

    // 2) dwconv7 + silu (x2)
    int convTot = B_ * L_ * CC_;
    k_dwconv_silu<<<(convTot + 255) / 256, 256, 0, stream>>>(zx + DI_, ZDIM, conv_w, conv_b, xf);
    k_dwconv_silu<<<(convTot + 255) / 256, 256, 0, stream>>>(xf, CC_, convb_w, convb_b, xb);

    // 3) CPB table, box corners, distance features
    k_table<<<4, 256, 0, stream>>>(cpb_w1, cpb_b1, cpb_w2, table);
    k_corners<<<1, 256, 0, stream>>>(query_xyz, query_size, query_angle, corners, Rm);
    k_dist<<<(B_ * L_ * K_) / 256, 256, 0, stream>>>(key_xyz, corners, Rm, table, dist);

    // 4) per-(b,l,k) scan coefficients
    k_bcdt<<<(B_ * L_ * K_) / 256, 256, 0, stream>>>(dist, zx, xf, xb, Wbc, Wdt, dt_bias, A_log,
                                                     dA, dBf, dBb, CmF, CmB);

    // 5) h0 = in_query @ Wq^T, transposed to [b][h][d][k]
    k_wmma_gemm_h0<<<dim3(DI_ / 16, (B_ * K_) / 16), 32, 0, stream>>>(in_query, Wq, h0t);

    // 6) bidirectional selective scan (32 independent scans)
    k_scan<<<32, 256, 0, stream>>>(dA, dBf, dBb, CmF, CmB, xf, xb, h0t, yF, yB, hF, hB);

    // 7) gate + RMS-norm, then out_key = gatedn @ Wok^T (exact N, unguarded)
    k_gate_norm<<<B_ * L_, 256, 0, stream>>>(yF, yB, xf, zx, Dp, key_norm_w, gatedn);
    k_wmma_gemm<false><<<dim3(D_ / 16, (B_ * L_) / 16), 32, 0, stream>>>(
        gatedn, Wok, out, B_ * L_, D_, DI_, DI_, DI_, D_);

    // 8) state LayerNorm, then out_query = stn @ Woq^T (exact N, unguarded)
    k_state_norm<<<B_ * K_, 256, 0, stream>>>(hF, hB, ln_w, ln_b, stn);
    k_wmma_gemm<false><<<dim3(D_ / 16, (B_ * K_) / 16), 32, 0, stream>>>(
        stn, Woq, out + (size_t)B_ * L_ * D_, B_ * K_, D_, DI_, DI_, DI_, D_);
}